// Initialize2_6399501271266
// MI455X (gfx1250) — compile-verified
//
#include <hip/hip_runtime.h>
#include <math.h>

// Problem constants (from the reference: x is [200, 3, 240, 320] float32 of byte values)
#define B_DIM   200
#define P_DIM   230400            // 3*240*320
#define TPB     256               // threads per block == pixels per block
#define NBLK    (P_DIM / TPB)     // 900 blocks
#define CHUNK   8                 // batch rows staged per async chunk
#define NCHUNK  (B_DIM / CHUNK)   // 25

// CDNA5 async global->LDS path (ASYNCcnt). Probe-confirmed present; param 0 is
// 'int __attribute__((address_space(1)))*' (non-const), per hipcc diagnostics.
#if defined(__HIP_DEVICE_COMPILE__) && defined(__gfx1250__) && \
    __has_builtin(__builtin_amdgcn_global_load_async_to_lds_b32) && \
    __has_builtin(__builtin_amdgcn_s_wait_asynccnt)
#define USE_ASYNC 1
#else
#define USE_ASYNC 0
#endif

__global__ __launch_bounds__(TPB) void mode_bgfg_kernel(const float* __restrict__ x,
                                                        float* __restrict__ out_bg,
                                                        float* __restrict__ out_fg) {
    // Per-thread byte-packed 256-bin histogram: 64 dwords per thread.
    // hist[i * TPB + tid]: lane -> bank = tid % 64 -> conflict-free in wave32.
    __shared__ unsigned int hist[64 * TPB];        // 64 KB
#if USE_ASYNC
    __shared__ float stage[2][CHUNK * TPB];        // 2 x 8 KB double buffer
#endif

    const int tid = threadIdx.x;
    const int p   = blockIdx.x * TPB + tid;
    const float* __restrict__ xp = x + p;          // column base, row stride = P_DIM

    // Zero own histogram column (no cross-thread sharing -> no barrier needed).
#pragma unroll
    for (int i = 0; i < 64; ++i) hist[i * TPB + tid] = 0u;

    // ---------------- Pass 1: build per-pixel histogram over the batch ----------------
#if USE_ASYNC
    typedef __attribute__((address_space(1))) int g_int;
    typedef __attribute__((address_space(3))) int lds_int;

    // Prologue: issue chunk 0.
#pragma unroll
    for (int bb = 0; bb < CHUNK; ++bb) {
        __builtin_amdgcn_global_load_async_to_lds_b32(
            (g_int*)(xp + (size_t)bb * P_DIM),
            (lds_int*)&stage[0][bb * TPB + tid],
            0, 0);
    }
    for (int c = 0; c < NCHUNK; ++c) {
        if (c + 1 < NCHUNK) {
#pragma unroll
            for (int bb = 0; bb < CHUNK; ++bb) {
                __builtin_amdgcn_global_load_async_to_lds_b32(
                    (g_int*)(xp + (size_t)((c + 1) * CHUNK + bb) * P_DIM),
                    (lds_int*)&stage[(c + 1) & 1][bb * TPB + tid],
                    0, 0);
            }
            // <= CHUNK outstanding -> chunk c (the older CHUNK ops) has landed in LDS.
            __builtin_amdgcn_s_wait_asynccnt(CHUNK);
        } else {
            __builtin_amdgcn_s_wait_asynccnt(0);
        }
#pragma unroll
        for (int bb = 0; bb < CHUNK; ++bb) {
            float    f = stage[c & 1][bb * TPB + tid];
            unsigned v = (unsigned)(int)f & 255u;
            // ds_add_u32, no return: fire-and-forget byte-lane increment (count <= 200 < 256).
            atomicAdd(&hist[(v >> 2) * TPB + tid], 1u << ((v & 3u) * 8u));
        }
    }
#else
    for (int b = 0; b < B_DIM; ++b) {
        float    f = xp[(size_t)b * P_DIM];
        unsigned v = (unsigned)(int)f & 255u;
        atomicAdd(&hist[(v >> 2) * TPB + tid], 1u << ((v & 3u) * 8u));
    }
#endif

    __syncthreads();   // also guarantees DScnt drained before the scan below

    // ---------------- Argmax over 256 bins (first max wins, scan ascending) ----------------
    int bestCnt = -1, bestBin = 0;
#pragma unroll
    for (int i = 0; i < 64; ++i) {
        unsigned w = hist[i * TPB + tid];
#pragma unroll
        for (int j = 0; j < 4; ++j) {
            int cnt = (int)((w >> (8 * j)) & 255u);
            if (cnt > bestCnt) { bestCnt = cnt; bestBin = i * 4 + j; }
        }
    }
    const float modef = (float)bestBin;

    // ---------------- Pass 2: bg/fg. x re-read is L2-hot; stores are non-temporal ----------------
    float* __restrict__ bgp = out_bg + p;
    float* __restrict__ fgp = out_fg + p;
    for (int b = 0; b < B_DIM; ++b) {
        const size_t off = (size_t)b * P_DIM;
        float f = xp[off];
        __builtin_nontemporal_store(modef, &bgp[off]);
        __builtin_nontemporal_store(fabsf(f - modef), &fgp[off]);
    }
}

extern "C" void kernel_launch(void* const* d_in, const int* in_sizes, int n_in,
                              void* d_out, int out_size, void* d_ws, size_t ws_size,
                              hipStream_t stream) {
    const float* x   = (const float*)d_in[0];
    float* out_bg    = (float*)d_out;
    float* out_fg    = out_bg + (size_t)B_DIM * P_DIM;
    mode_bgfg_kernel<<<NBLK, TPB, 0, stream>>>(x, out_bg, out_fg);
}